// SinkhornDistance_3272765079554
// MI455X (gfx1250) — compile-verified
//
#include <hip/hip_runtime.h>
#include <math.h>

#define BATCH 8
#define NP 2048
#define ND 64
#define EPS 0.1f
#define INV_EPS 10.0f
#define LOG_MU (-7.6246189861593985f)   /* -log(2048) == log_nu */
#define THRESH 0.01f
#define HALF_ITERS 25                   /* MAX_ITER/2 scan steps */

typedef __attribute__((ext_vector_type(2))) float v2f;
typedef __attribute__((ext_vector_type(4))) float v4f;
typedef __attribute__((ext_vector_type(8))) float v8f;

// ---------------------------------------------------------------------------
// init: zero u, v, err accumulator, done flag in workspace
// ---------------------------------------------------------------------------
__global__ void init_kernel(float* ws, int n) {
    int t = blockIdx.x * blockDim.x + threadIdx.x;
    if (t < n) ws[t] = 0.0f;
}

// ---------------------------------------------------------------------------
// Cost matrix: C[b,i,j] = ||x_i||^2 + ||y_j||^2 - 2 * dot(x_i, y_j)
// 64x64 output tile per 256-thread block; GEMM part via V_WMMA_F32_16X16X4_F32
// ---------------------------------------------------------------------------
__global__ __launch_bounds__(256) void cost_kernel(const float* __restrict__ x,
                                                   const float* __restrict__ y,
                                                   float* __restrict__ C) {
    __shared__ float lx[64 * 68];   // padded stride 68 to avoid bank conflicts
    __shared__ float ly[64 * 68];
    __shared__ float lxx[64];
    __shared__ float lyy[64];

    const int b  = blockIdx.z;
    const int ib = blockIdx.y * 64;
    const int jb = blockIdx.x * 64;
    const int t  = threadIdx.x;

    const float* xb = x + ((size_t)b * NP + ib) * ND;
    const float* yb = y + ((size_t)b * NP + jb) * ND;

    // cooperative load of 64x64 x-tile and y-tile (float4, fully coalesced)
    #pragma unroll
    for (int r = 0; r < 4; ++r) {
        int idx = t + r * 256;          // float4 index, 1024 per tile
        int row = idx >> 4;             // 16 float4 per 64-float row
        int c4  = (idx & 15) << 2;
        v4f vx = *(const v4f*)(xb + row * ND + c4);
        v4f vy = *(const v4f*)(yb + row * ND + c4);
        *(v4f*)(&lx[row * 68 + c4]) = vx;
        *(v4f*)(&ly[row * 68 + c4]) = vy;
    }
    __syncthreads();

    // row squared norms from LDS
    if (t < 64) {
        float s = 0.0f;
        #pragma unroll 8
        for (int k = 0; k < 64; ++k) { float a = lx[t * 68 + k]; s += a * a; }
        lxx[t] = s;
    } else if (t < 128) {
        int r = t - 64;
        float s = 0.0f;
        #pragma unroll 8
        for (int k = 0; k < 64; ++k) { float a = ly[r * 68 + k]; s += a * a; }
        lyy[r] = s;
    }
    __syncthreads();

    const int wave = t >> 5;
    const int lane = t & 31;
    const int m    = lane & 15;    // M (for A) / N (for B) index within tile
    const int half = lane >> 4;    // K sub-pair selector per 16x16x4 layout

    #pragma unroll
    for (int tt = 0; tt < 2; ++tt) {
        const int tile = wave + tt * 8;        // 16 wave-tiles, 8 waves x 2
        const int ti = tile >> 2;
        const int tj = tile & 3;
        const float* arow = &lx[(ti * 16 + m) * 68];
        const float* brow = &ly[(tj * 16 + m) * 68];

        v8f acc = {};
        #pragma unroll
        for (int ks = 0; ks < 16; ++ks) {      // K = 64 in steps of 4
            const int k0 = ks * 4 + half * 2;
            v2f a  = *(const v2f*)(arow + k0);
            v2f bm = *(const v2f*)(brow + k0);
            acc = __builtin_amdgcn_wmma_f32_16x16x4_f32(
                false, a, false, bm, (short)0, acc, false, false);
        }

        // C/D layout: VGPR v -> row M = v + 8*half ; col N = lane & 15
        const float yyn  = lyy[tj * 16 + m];
        const size_t base = ((size_t)b * NP + (ib + ti * 16)) * NP
                          + (size_t)(jb + tj * 16 + m);
        #pragma unroll
        for (int v = 0; v < 8; ++v) {
            const int rm = v + half * 8;
            C[base + (size_t)rm * NP] =
                lxx[ti * 16 + rm] + yyn - 2.0f * acc[v];
        }
    }
}

// ---------------------------------------------------------------------------
// u-update: one wave32 per row. u_new_i = eps*(log_mu - lse_j((v_j - C_ij)/eps))
// ---------------------------------------------------------------------------
__global__ __launch_bounds__(256) void u_kernel(const float* __restrict__ C,
                                                float* __restrict__ u,
                                                const float* __restrict__ v,
                                                float* __restrict__ err,
                                                const unsigned* __restrict__ done) {
    if (*done) return;                      // frozen carry: u stays, err stays 0
    const int row  = (blockIdx.x * blockDim.x + threadIdx.x) >> 5;
    const int lane = threadIdx.x & 31;
    if (row >= BATCH * NP) return;
    const int b = row / NP;
    const float* crow = C + (size_t)row * NP;
    const float* vb   = v + (size_t)b * NP;

    float mx = -INFINITY, s = 0.0f;         // streaming logsumexp
    for (int j = lane; j < NP; j += 32) {
        float tv = (vb[j] - crow[j]) * INV_EPS;
        if (tv > mx) { s = s * __expf(mx - tv) + 1.0f; mx = tv; }
        else         { s += __expf(tv - mx); }
    }
    #pragma unroll
    for (int off = 16; off; off >>= 1) {    // wave32 (m,s) merge
        float m2 = __shfl_xor(mx, off, 32);
        float s2 = __shfl_xor(s,  off, 32);
        if (m2 > mx) { s = s * __expf(mx - m2) + s2; mx = m2; }
        else         { s += s2 * __expf(m2 - mx); }
    }
    if (lane == 0) {
        const float lse = mx + __logf(s);
        const float un  = EPS * (LOG_MU - lse);
        const float du  = fabsf(un - u[row]);
        u[row] = un;
        atomicAdd(err, du);
    }
}

// ---------------------------------------------------------------------------
// err fold: err = sum|du| / BATCH ; done |= err < THRESH ; reset accumulator
// ---------------------------------------------------------------------------
__global__ void err_kernel(float* err, unsigned* done) {
    const float e = *err * (1.0f / (float)BATCH);
    if (e < THRESH) *done = 1u;
    *err = 0.0f;
}

// ---------------------------------------------------------------------------
// v-update: one thread per column (coalesced across j).
// v_new_j = eps*(log_nu - lse_i((u_i - C_ij)/eps))
// ---------------------------------------------------------------------------
__global__ __launch_bounds__(256) void v_kernel(const float* __restrict__ C,
                                                const float* __restrict__ u,
                                                float* __restrict__ v,
                                                const unsigned* __restrict__ done) {
    if (*done) return;
    const int tid = blockIdx.x * blockDim.x + threadIdx.x;
    if (tid >= BATCH * NP) return;
    const int b = tid / NP;
    const int j = tid % NP;
    const float* col = C + (size_t)b * NP * NP + j;
    const float* ub  = u + (size_t)b * NP;

    float mx = -INFINITY, s = 0.0f;
    for (int i = 0; i < NP; ++i) {
        float tv = (ub[i] - col[(size_t)i * NP]) * INV_EPS;
        if (tv > mx) { s = s * __expf(mx - tv) + 1.0f; mx = tv; }
        else         { s += __expf(tv - mx); }
    }
    v[tid] = EPS * (LOG_MU - (mx + __logf(s)));
}

// ---------------------------------------------------------------------------
// pi = exp((-C + u_i + v_j)/eps), float4 per thread
// ---------------------------------------------------------------------------
__global__ __launch_bounds__(256) void pi_kernel(const float* __restrict__ C,
                                                 const float* __restrict__ u,
                                                 const float* __restrict__ v,
                                                 float* __restrict__ pi) {
    const size_t idx = ((size_t)blockIdx.x * blockDim.x + threadIdx.x) * 4;
    const size_t pp  = (size_t)NP * NP;
    const int b = (int)(idx / pp);
    const size_t rem = idx - (size_t)b * pp;
    const int i = (int)(rem / NP);
    const int j = (int)(rem % NP);
    const float ui = u[b * NP + i];
    const float* vb = v + b * NP;
    v4f c = *(const v4f*)(C + idx);
    v4f o;
    o.x = __expf((ui + vb[j + 0] - c.x) * INV_EPS);
    o.y = __expf((ui + vb[j + 1] - c.y) * INV_EPS);
    o.z = __expf((ui + vb[j + 2] - c.z) * INV_EPS);
    o.w = __expf((ui + vb[j + 3] - c.w) * INV_EPS);
    *(v4f*)(pi + idx) = o;
}

// ---------------------------------------------------------------------------
// copy u, v into the output tail
// ---------------------------------------------------------------------------
__global__ void uv_copy_kernel(const float* __restrict__ u,
                               const float* __restrict__ v,
                               float* __restrict__ out) {
    const int t = blockIdx.x * blockDim.x + threadIdx.x;
    const int n = BATCH * NP;
    if (t < n)            out[t]     = u[t];
    else if (t < 2 * n)   out[t]     = v[t - n];
}

extern "C" void kernel_launch(void* const* d_in, const int* in_sizes, int n_in,
                              void* d_out, int out_size, void* d_ws, size_t ws_size,
                              hipStream_t stream) {
    const float* x = (const float*)d_in[0];
    const float* y = (const float*)d_in[1];
    float* out = (float*)d_out;

    const size_t BPP = (size_t)BATCH * NP * NP;
    float* pi_out = out;             // (8,2048,2048)
    float* C_out  = out + BPP;       // (8,2048,2048)
    float* uv_out = out + 2 * BPP;   // u (8,2048) then v (8,2048)

    float*    u    = (float*)d_ws;                 // 16384
    float*    v    = u + BATCH * NP;               // 16384
    float*    err  = v + BATCH * NP;               // 1
    unsigned* done = (unsigned*)(err + 1);         // 1

    // zero u, v, err, done (re-init every call -> deterministic)
    const int ws_n = 2 * BATCH * NP + 2;
    init_kernel<<<(ws_n + 255) / 256, 256, 0, stream>>>((float*)d_ws, ws_n);

    // cost matrix with fp32 WMMA
    dim3 cg(NP / 64, NP / 64, BATCH);
    cost_kernel<<<cg, 256, 0, stream>>>(x, y, C_out);

    // Sinkhorn scan: 25 x (u-update, err/done fold, v-update)
    const int rows = BATCH * NP;                   // 16384
    const int ublocks = (rows * 32) / 256;         // one wave per row
    const int vblocks = rows / 256;                // one thread per column
    for (int it = 0; it < HALF_ITERS; ++it) {
        u_kernel<<<ublocks, 256, 0, stream>>>(C_out, u, v, err, done);
        err_kernel<<<1, 1, 0, stream>>>(err, done);
        v_kernel<<<vblocks, 256, 0, stream>>>(C_out, u, v, done);
    }

    // transport plan + u/v outputs
    pi_kernel<<<(unsigned)(BPP / 4 / 256), 256, 0, stream>>>(C_out, u, v, pi_out);
    uv_copy_kernel<<<(2 * rows + 255) / 256, 256, 0, stream>>>(u, v, uv_out);
}